// Project_23682449670327
// MI455X (gfx1250) — compile-verified
//
#include <hip/hip_runtime.h>
#include <hip/hip_bf16.h>
#include <stdint.h>

// ---------------- problem geometry (must match the reference) ----------------
#define NXI 256
#define NYI 256
#define EVENT_NUM 100000
#define NSAMP 512
#define LANES_PER_EVENT 4                       // 4 lanes march one LOR
#define SAMP_PER_LANE (NSAMP / LANES_PER_EVENT) // 128 samples per lane
#define BLOCK 1024                              // 32 wave32s = one full WGP
#define LDS_BYTES (NXI * NYI * 4)               // 256 KB image tile in LDS

// sigma = TIME_RES_PS * C * 0.5 / FWHM_TO_SIGMA
#define SIGMA_F   ((float)(300.0 * 0.299792458 * 0.5 / 2.3548200450309493))
#define INVSIG_F  (1.0f / SIGMA_F)
#define INVNORM_F ((float)(0.3989422804014327 / (300.0 * 0.299792458 * 0.5 / 2.3548200450309493)))
#define HALF_C_F  0.149896229f                  // 0.5 * mm per ps

typedef unsigned int uint32x4 __attribute__((ext_vector_type(4)));
typedef int          int32x4  __attribute__((ext_vector_type(4)));
typedef int          int32x8  __attribute__((ext_vector_type(8)));

__global__ __launch_bounds__(BLOCK) void pet_tof_fwd_kernel(
    const float* __restrict__ image,
    const float* __restrict__ tof,
    const float* __restrict__ x1l, const float* __restrict__ y1l,
    const float* __restrict__ x1r, const float* __restrict__ y1r,
    const float* __restrict__ x2l, const float* __restrict__ y2l,
    const float* __restrict__ x2r, const float* __restrict__ y2r,
    float* __restrict__ out)
{
    extern __shared__ float simg[];   // 256*256 f32 = 256 KB (< 320 KB WGP LDS)

    // ---- stage whole image Global -> LDS with ONE Tensor-Data-Mover op ----
    // Only wave 0 issues the descriptor (TDM ignores EXEC; per-wave issue).
    if (threadIdx.x < 32) {
        unsigned long long ga = (unsigned long long)(uintptr_t)image;
        unsigned int lds_off  = (unsigned int)(uintptr_t)(&simg[0]); // low 32b of flat addr = LDS byte offset

        uint32x4 g0;
        g0.x = 1u;                                  // count=1 (valid), user mode, no gather
        g0.y = lds_off;                             // lds_addr
        g0.z = (unsigned int)(ga & 0xFFFFFFFFull);  // global_addr[31:0]
        g0.w = (unsigned int)((ga >> 32) & 0x1FFFFFFull) | 0x80000000u; // addr[56:32] | type=2

        int32x8 g1;
        g1[0] = 0x00020000;          // workgroup_mask=0 (not in cluster), data_size=2 -> 4 bytes
        g1[1] = (int)(NYI << 16);    // tensor_dim0[15:0] = 256 (fastest dim = iy)
        g1[2] = (int)(NXI << 16);    // tensor_dim0 hi = 0 ; tensor_dim1[15:0] = 256
        g1[3] = (int)(NYI << 16);    // tensor_dim1 hi = 0 ; tile_dim0 = 256
        g1[4] = NXI;                 // tile_dim1 = 256 ; tile_dim2 = 0
        g1[5] = NYI;                 // tensor_dim0_stride[31:0] = 256
        g1[6] = 0;                   // stride0 hi = 0 ; tensor_dim1_stride[15:0] = 0
        g1[7] = 1;                   // tensor_dim1_stride[47:16] = 1 -> 65536

        int32x4 gz = {0, 0, 0, 0};   // groups 2/3 unused (2D tile)
#if defined(__clang_major__) && (__clang_major__ >= 23)
        int32x8 gz8 = {0, 0, 0, 0, 0, 0, 0, 0};
        __builtin_amdgcn_tensor_load_to_lds(g0, g1, gz, gz, gz8, 0);
#else
        __builtin_amdgcn_tensor_load_to_lds(g0, g1, gz, gz, 0);
#endif
        __builtin_amdgcn_s_wait_tensorcnt(0);   // TENSORcnt -> 0: DMA complete
    }
    __syncthreads();                            // all 32 waves see the tile

    // ------------------- per-event TOF ray marching -------------------
    const int tid = blockIdx.x * BLOCK + threadIdx.x;
    const int e   = tid >> 2;          // event index (4 lanes per event)
    const int sub = tid & 3;           // which quarter of the 512 samples

    float acc = 0.0f;
    float stepLen = 0.0f;

    if (e < EVENT_NUM) {
        float p1x = 0.5f * (x1l[e] + x1r[e]);
        float p1y = 0.5f * (y1l[e] + y1r[e]);
        float p2x = 0.5f * (x2l[e] + x2r[e]);
        float p2y = 0.5f * (y2l[e] + y2r[e]);
        float dxv = p2x - p1x, dyv = p2y - p1y;
        float L   = sqrtf(fmaf(dxv, dxv, dyv * dyv)) + 1e-8f;
        float invL = 1.0f / L;
        float ux = dxv * invL, uy = dyv * invL;
        stepLen  = L * (1.0f / (float)NSAMP);
        float sc = fmaf(HALF_C_F, tof[e], 0.5f * L);

        // this lane covers samples [sub*128, sub*128+128)
        float t0  = ((float)(sub * SAMP_PER_LANE) + 0.5f) * stepLen;
        // world(mm) -> pixel: p/2 + 127.5 ; advance incrementally
        float px  = fmaf(t0, ux, p1x) * 0.5f + 127.5f;
        float py  = fmaf(t0, uy, p1y) * 0.5f + 127.5f;
        float dpx = stepLen * ux * 0.5f;
        float dpy = stepLen * uy * 0.5f;
        float z   = (t0 - sc) * INVSIG_F;
        float dz  = stepLen * INVSIG_F;

#pragma unroll 4
        for (int i = 0; i < SAMP_PER_LANE; ++i) {
            float fxp = floorf(px), fyp = floorf(py);
            int   ix0 = (int)fxp,   iy0 = (int)fyp;
            float fx  = px - fxp,   fy  = py - fyp;

            // y-validity folded into the y-interp weights (kills row aliasing);
            // x-OOB addresses fall outside the 256 KB LDS allocation and
            // hardware-return zero (CDNA5 LDS OOB semantics) == reference pad.
            float wy0 = ((unsigned)iy0       < (unsigned)NYI) ? (1.0f - fy) : 0.0f;
            float wy1 = ((unsigned)(iy0 + 1) < (unsigned)NYI) ? fy          : 0.0f;

            int base0 = (ix0 << 8) + iy0;     // row ix0
            int base1 = base0 + NYI;          // row ix0+1
            float v00 = simg[base0];
            float v01 = simg[base0 + 1];
            float v10 = simg[base1];
            float v11 = simg[base1 + 1];

            float row0 = fmaf(fx, v10 - v00, v00);   // lerp along x at iy0
            float row1 = fmaf(fx, v11 - v01, v01);   // lerp along x at iy0+1
            float val  = fmaf(row1, wy1, row0 * wy0);

            float w = __expf(-0.5f * z * z);         // v_exp_f32 (trans, co-exec)
            acc = fmaf(val, w, acc);

            px += dpx; py += dpy; z += dz;
        }
    }

    // ---- reduce the 4 lanes of one event (wave32 quad shuffle) ----
    acc += __shfl_xor(acc, 1, 32);
    acc += __shfl_xor(acc, 2, 32);
    if (sub == 0 && e < EVENT_NUM)
        out[e] = acc * stepLen * INVNORM_F;
}

extern "C" void kernel_launch(void* const* d_in, const int* in_sizes, int n_in,
                              void* d_out, int out_size, void* d_ws, size_t ws_size,
                              hipStream_t stream) {
    (void)in_sizes; (void)n_in; (void)d_ws; (void)ws_size; (void)out_size;
    const float* image = (const float*)d_in[0];
    const float* tof   = (const float*)d_in[1];
    const float* x1l   = (const float*)d_in[2];
    const float* y1l   = (const float*)d_in[3];
    const float* x1r   = (const float*)d_in[4];
    const float* y1r   = (const float*)d_in[5];
    const float* x2l   = (const float*)d_in[6];
    const float* y2l   = (const float*)d_in[7];
    const float* x2r   = (const float*)d_in[8];
    const float* y2r   = (const float*)d_in[9];
    float* out = (float*)d_out;

    const int total_threads = EVENT_NUM * LANES_PER_EVENT;        // 400000
    const int blocks = (total_threads + BLOCK - 1) / BLOCK;       // 391

    pet_tof_fwd_kernel<<<blocks, BLOCK, LDS_BYTES, stream>>>(
        image, tof, x1l, y1l, x1r, y1r, x2l, y2l, x2r, y2r, out);
}